// GATFraudDetector_7584912245132
// MI455X (gfx1250) — compile-verified
//
#include <hip/hip_runtime.h>
#include <limits.h>

// ---------------- Problem constants ----------------
#define NNODE 50000
#define NEDGE 800000
#define NPAIR 200000
#define IN_DIM 32
#define HID 64
#define HEADS 4
#define DDIM 256              // HID*HEADS
#define ETOT (NEDGE + NNODE)  // edges + self loops
#define ND ((size_t)NNODE * DDIM)
#define NH (NNODE * HEADS)
#define NEG_SLOPE 0.2f

typedef __attribute__((ext_vector_type(2))) float v2f;
typedef __attribute__((ext_vector_type(8))) float v8f;

// ---------------- helpers ----------------
__device__ __forceinline__ int enc_f(float f) {
  int i = __float_as_int(f);
  return i >= 0 ? i : (i ^ 0x7fffffff);   // monotone int encoding of float order
}
__device__ __forceinline__ float dec_f(int k) {
  return __int_as_float(k >= 0 ? k : (k ^ 0x7fffffff));
}
__device__ __forceinline__ float leaky(float v) {
  return v > 0.f ? v : NEG_SLOPE * v;
}

__global__ void fill_f32_kernel(float* __restrict__ p, float v, int n) {
  int t = blockIdx.x * blockDim.x + threadIdx.x;
  if (t < n) p[t] = v;
}
__global__ void fill_i32_kernel(int* __restrict__ p, int v, int n) {
  int t = blockIdx.x * blockDim.x + threadIdx.x;
  if (t < n) p[t] = v;
}

// ---------------- fp32 WMMA GEMM: Y[M x Nout] = X[M x K] @ W[K x Nout] ------
// One wave computes a 16x64 strip (4 accumulator tiles) via
// V_WMMA_F32_16X16X4_F32; the A fragment is reused across the 4 tiles.
// A frag (16x4): lane l -> M = l&15, K = (l>>4)*2 + v  (v = vgpr 0/1)
// B frag (4x16): lane l -> N = l&15, K = (l>>4)*2 + v
// C/D (16x16):   vgpr v -> M = v + 8*(l>>4), N = l&15
__global__ void __launch_bounds__(32)
gemm_wmma_f32_kernel(const float* __restrict__ X, const float* __restrict__ W,
                     float* __restrict__ Y, int K, int Nout) {
  const int m0 = blockIdx.x * 16;
  const int n0 = blockIdx.y * 64;    // 4 x 16 output tiles per wave
  const int l  = threadIdx.x;        // 0..31, full wave always active
  const int ml = l & 15;
  const int kb = (l >> 4) * 2;

  const float* __restrict__ xrow = X + (size_t)(m0 + ml) * K;
  v8f acc0 = {}, acc1 = {}, acc2 = {}, acc3 = {};
  for (int k0 = 0; k0 < K; k0 += 4) {
    v2f a;
    a.x = xrow[k0 + kb];
    a.y = xrow[k0 + kb + 1];
    const float* __restrict__ w0 = W + (size_t)(k0 + kb) * Nout + ml;
    const float* __restrict__ w1 = W + (size_t)(k0 + kb + 1) * Nout + ml;
    v2f b0 = { w0[n0 +  0], w1[n0 +  0] };
    v2f b1 = { w0[n0 + 16], w1[n0 + 16] };
    v2f b2 = { w0[n0 + 32], w1[n0 + 32] };
    v2f b3 = { w0[n0 + 48], w1[n0 + 48] };
    acc0 = __builtin_amdgcn_wmma_f32_16x16x4_f32(false, a, false, b0,
                                                 (short)0, acc0, false, false);
    acc1 = __builtin_amdgcn_wmma_f32_16x16x4_f32(false, a, false, b1,
                                                 (short)0, acc1, false, false);
    acc2 = __builtin_amdgcn_wmma_f32_16x16x4_f32(false, a, false, b2,
                                                 (short)0, acc2, false, false);
    acc3 = __builtin_amdgcn_wmma_f32_16x16x4_f32(false, a, false, b3,
                                                 (short)0, acc3, false, false);
  }
  const int rbase = m0 + (l >> 4) * 8;
#pragma unroll
  for (int v = 0; v < 8; ++v) {
    float* yr = Y + (size_t)(rbase + v) * Nout + ml;
    yr[n0 +  0] = acc0[v];
    yr[n0 + 16] = acc1[v];
    yr[n0 + 32] = acc2[v];
    yr[n0 + 48] = acc3[v];
  }
}

// ---------------- per-(node,head) attention dots -------------------------
// wave w handles pair p = w (p = node*4 + head); h row slice is exactly
// 64 consecutive floats at h + p*64.
__global__ void alphas_kernel(const float* __restrict__ h,
                              const float* __restrict__ asrc,
                              const float* __restrict__ adst,
                              float* __restrict__ out_s, float* __restrict__ out_d,
                              int total) {
  int wave = (blockIdx.x * blockDim.x + threadIdx.x) >> 5;
  int lane = threadIdx.x & 31;
  if (wave >= total) return;
  int head = wave & 3;
  const float* hp = h + (size_t)wave * 64;
  const float* ap = asrc + head * 64;
  const float* dp = adst + head * 64;
  float s = hp[lane] * ap[lane] + hp[lane + 32] * ap[lane + 32];
  float d = hp[lane] * dp[lane] + hp[lane + 32] * dp[lane + 32];
#pragma unroll
  for (int off = 16; off; off >>= 1) {
    s += __shfl_xor(s, off, 32);
    d += __shfl_xor(d, off, 32);
  }
  if (lane == 0) { out_s[wave] = s; out_d[wave] = d; }
}

__device__ __forceinline__ void edge_sd(const int* __restrict__ ei, int e,
                                        int& s, int& d) {
  if (e < NEDGE) { s = ei[e]; d = ei[NEDGE + e]; }
  else           { s = d = e - NEDGE; }   // self loops
}

// ---------------- segment max over edges (per dst,head) ------------------
__global__ void edge_max_kernel(const int* __restrict__ ei,
                                const float* __restrict__ as_,
                                const float* __restrict__ ad_,
                                int* __restrict__ m) {
  int t = blockIdx.x * blockDim.x + threadIdx.x;
  if (t >= ETOT * HEADS) return;
  int e = t >> 2, h = t & 3;
  int s, d; edge_sd(ei, e, s, d);
  float v = leaky(as_[s * 4 + h] + ad_[d * 4 + h]);
  atomicMax(&m[d * 4 + h], enc_f(v));
}

// ---------------- segment exp-sum ---------------------------------------
__global__ void edge_expsum_kernel(const int* __restrict__ ei,
                                   const float* __restrict__ as_,
                                   const float* __restrict__ ad_,
                                   const int* __restrict__ m,
                                   float* __restrict__ den) {
  int t = blockIdx.x * blockDim.x + threadIdx.x;
  if (t >= ETOT * HEADS) return;
  int e = t >> 2, h = t & 3;
  int s, d; edge_sd(ei, e, s, d);
  float v = leaky(as_[s * 4 + h] + ad_[d * 4 + h]);
  atomicAdd(&den[d * 4 + h], __expf(v - dec_f(m[d * 4 + h])));
}

// ---------------- weighted message scatter (wave per edge) ---------------
__global__ void edge_agg_kernel(const int* __restrict__ ei,
                                const float* __restrict__ as_,
                                const float* __restrict__ ad_,
                                const int* __restrict__ m,
                                const float* __restrict__ den,
                                const float* __restrict__ hin,
                                float* __restrict__ out) {
  int wave = (blockIdx.x * blockDim.x + threadIdx.x) >> 5;
  int lane = threadIdx.x & 31;
  if (wave >= ETOT) return;
  int s, d; edge_sd(ei, wave, s, d);
  float al[HEADS];
#pragma unroll
  for (int h = 0; h < HEADS; ++h) {
    float v  = leaky(as_[s * 4 + h] + ad_[d * 4 + h]);
    float ex = __expf(v - dec_f(m[d * 4 + h]));
    al[h] = ex / (den[d * 4 + h] + 1e-16f);
  }
  const float* __restrict__ hs = hin + (size_t)s * DDIM;
  float* __restrict__ od = out + (size_t)d * DDIM;
#pragma unroll
  for (int k = 0; k < 8; ++k) {            // channel c = lane + 32k, head = k>>1
    int c = lane + 32 * k;
    atomicAdd(&od[c], hs[c] * al[k >> 1]);
  }
}

// ---------------- bias + relu in place -----------------------------------
__global__ void relu_bias_kernel(float* __restrict__ buf,
                                 const float* __restrict__ b, int total) {
  int t = blockIdx.x * blockDim.x + threadIdx.x;
  if (t >= total) return;
  float v = buf[t] + b[t & (DDIM - 1)];
  buf[t] = v > 0.f ? v : 0.f;
}

// ---------------- pair scoring head (wave per pair) ----------------------
__global__ void head_kernel(const float* __restrict__ h,
                            const int* __restrict__ pairs,
                            const float* __restrict__ w,
                            const float* __restrict__ hb,
                            float* __restrict__ out, int total) {
  int wave = (blockIdx.x * blockDim.x + threadIdx.x) >> 5;
  int lane = threadIdx.x & 31;
  if (wave >= total) return;
  int a = pairs[wave * 2 + 0];
  int b = pairs[wave * 2 + 1];
  const float* ha = h + (size_t)a * DDIM;
  const float* hbp = h + (size_t)b * DDIM;
  float s = 0.f;
#pragma unroll
  for (int k = 0; k < 8; ++k) {
    int c = lane + 32 * k;
    s += ha[c] * w[c] + hbp[c] * w[DDIM + c];
  }
#pragma unroll
  for (int off = 16; off; off >>= 1) s += __shfl_xor(s, off, 32);
  if (lane == 0) out[wave] = s + hb[0];
}

// =========================================================================
extern "C" void kernel_launch(void* const* d_in, const int* in_sizes, int n_in,
                              void* d_out, int out_size, void* d_ws, size_t ws_size,
                              hipStream_t stream) {
  (void)in_sizes; (void)n_in; (void)out_size; (void)ws_size;
  const float* x        = (const float*)d_in[0];
  const int*   ei       = (const int*)  d_in[1];
  const int*   pairs    = (const int*)  d_in[2];
  const float* W1       = (const float*)d_in[3];
  const float* att_s1   = (const float*)d_in[4];
  const float* att_d1   = (const float*)d_in[5];
  const float* b1       = (const float*)d_in[6];
  const float* W2       = (const float*)d_in[7];
  const float* att_s2   = (const float*)d_in[8];
  const float* att_d2   = (const float*)d_in[9];
  const float* b2       = (const float*)d_in[10];
  const float* head_w   = (const float*)d_in[11];
  const float* head_b   = (const float*)d_in[12];
  float* out = (float*)d_out;

  float* bufA = (float*)d_ws;          // h_lin (GEMM output, both layers)
  float* bufB = bufA + ND;             // agg1 target / h1 / h2
  float* as_  = bufB + ND;             // [N,H] alpha_src
  float* ad_  = as_ + NH;              // [N,H] alpha_dst
  int*   m_   = (int*)(ad_ + NH);      // [N,H] encoded segment max
  float* den  = (float*)(m_ + NH);     // [N,H] softmax denom

  const int TB = 256;
  const int gNH   = (NH * 32 + TB - 1) / TB;        // wave per (node,head)
  const int gEh   = (ETOT * HEADS + TB - 1) / TB;   // thread per (edge,head)
  const int gEw   = ((size_t)ETOT * 32 + TB - 1) / TB; // wave per edge
  const int gND   = ((int)ND + TB - 1) / TB;
  const int gNHf  = (NH + TB - 1) / TB;
  const int gP    = (NPAIR * 32 + TB - 1) / TB;
  dim3 gemmGrid(NNODE / 16, DDIM / 64);             // 16x64 strip per wave

  // ---------------- layer 1 ----------------
  gemm_wmma_f32_kernel<<<gemmGrid, 32, 0, stream>>>(x, W1, bufA, IN_DIM, DDIM);
  alphas_kernel<<<gNH, TB, 0, stream>>>(bufA, att_s1, att_d1, as_, ad_, NH);
  fill_i32_kernel<<<gNHf, TB, 0, stream>>>(m_, INT_MIN, NH);
  fill_f32_kernel<<<gNHf, TB, 0, stream>>>(den, 0.f, NH);
  fill_f32_kernel<<<gND, TB, 0, stream>>>(bufB, 0.f, (int)ND);
  edge_max_kernel<<<gEh, TB, 0, stream>>>(ei, as_, ad_, m_);
  edge_expsum_kernel<<<gEh, TB, 0, stream>>>(ei, as_, ad_, m_, den);
  edge_agg_kernel<<<gEw, TB, 0, stream>>>(ei, as_, ad_, m_, den, bufA, bufB);
  relu_bias_kernel<<<gND, TB, 0, stream>>>(bufB, b1, (int)ND);   // bufB = h1

  // ---------------- layer 2 ----------------
  gemm_wmma_f32_kernel<<<gemmGrid, 32, 0, stream>>>(bufB, W2, bufA, DDIM, DDIM);
  alphas_kernel<<<gNH, TB, 0, stream>>>(bufA, att_s2, att_d2, as_, ad_, NH);
  fill_i32_kernel<<<gNHf, TB, 0, stream>>>(m_, INT_MIN, NH);
  fill_f32_kernel<<<gNHf, TB, 0, stream>>>(den, 0.f, NH);
  fill_f32_kernel<<<gND, TB, 0, stream>>>(bufB, 0.f, (int)ND);
  edge_max_kernel<<<gEh, TB, 0, stream>>>(ei, as_, ad_, m_);
  edge_expsum_kernel<<<gEh, TB, 0, stream>>>(ei, as_, ad_, m_, den);
  edge_agg_kernel<<<gEw, TB, 0, stream>>>(ei, as_, ad_, m_, den, bufA, bufB);
  relu_bias_kernel<<<gND, TB, 0, stream>>>(bufB, b2, (int)ND);   // bufB = h2

  // ---------------- pair head ----------------
  head_kernel<<<gP, TB, 0, stream>>>(bufB, pairs, head_w, head_b, out, NPAIR);
}